// StructureModule_36172214566968
// MI455X (gfx1250) — compile-verified
//
#include <hip/hip_runtime.h>
#include <math.h>

typedef float v2f __attribute__((ext_vector_type(2)));
typedef float v8f __attribute__((ext_vector_type(8)));

namespace {
constexpr int N   = 768;
constexpr int H   = 12;
constexpr int SQK = 16;
constexpr int SV  = 16;
constexpr int PQK = 4;
constexpr int PV  = 8;
constexpr int C1  = 384;
constexpr int C2  = 128;
constexpr int VCW = 48;                       // packed value width (40 padded to 48)
constexpr int FIN = H * (C2 + SV + 4 * PV);   // 2112
constexpr float INV_SQRT3 = 0.5773502691896258f;
}

// ---------------------------------------------------------------------------
// V_WMMA_F32_16X16X4_F32 wrapper.
// A: 16x4 f32 (2 VGPR/lane): lane L -> m=L&15, k-half=(L>>4)*2.
// B: 4x16 f32:               lane L -> n=L&15, k-half=(L>>4)*2.
// C/D 16x16 f32: VGPR j, lanes 0-15 -> (M=j, N=lane); 16-31 -> (M=j+8, N=lane-16)
// ---------------------------------------------------------------------------
__device__ __forceinline__ v8f wmma4(v2f a, v2f b, v8f c) {
  return __builtin_amdgcn_wmma_f32_16x16x4_f32(false, a, false, b,
                                               (short)0, c, false, false);
}

__device__ __forceinline__ v2f load2(const float* p) {
  return *(const v2f*)p;                      // 8B-aligned by construction
}

__device__ __forceinline__ float softplusf(float x) {
  return (x > 20.f) ? x : log1pf(expf(x));
}

// ---------------------------------------------------------------------------
// Generic f32 WMMA GEMM: C[M,Nc] = A[M,K] * B[K,Nc] (+bias), row-major.
// Each wave computes a 16 x (16*NT) tile: one A fragment feeds NT WMMAs.
// All inner-loop loads are unconditional (edge columns are clamped and the
// loaded values multiplied by a 0/1 mask) so EXEC never toggles in the loop.
// M % 16 == 0, K % 4 == 0, lda even.
// ---------------------------------------------------------------------------
template <int NT>
__global__ void gemm16_kernel(const float* __restrict__ A,
                              const float* __restrict__ B,
                              const float* __restrict__ bias,
                              float* __restrict__ C,
                              int M, int Nc, int K,
                              int lda, int ldb, int ldc) {
  const int wave = (blockIdx.x * blockDim.x + threadIdx.x) >> 5;
  const int lane = threadIdx.x & 31;
  const int mtiles  = M >> 4;
  const int nblocks = (Nc + 16 * NT - 1) / (16 * NT);
  if (wave >= mtiles * nblocks) return;
  const int mt = wave % mtiles;
  const int nb = wave / mtiles;
  const int row0 = mt << 4;
  const int am   = lane & 15;
  const int kh   = (lane >> 4) << 1;

  // Per-lane, loop-invariant column offsets + masks for the NT subtiles.
  int   bo0[NT], bo1[NT];
  float cm[NT];
  int   cols[NT];
#pragma unroll
  for (int t = 0; t < NT; ++t) {
    const int c = nb * 16 * NT + (t << 4) + (lane & 15);
    cols[t] = c;
    cm[t]   = (c < Nc) ? 1.f : 0.f;
    const int cc = (c < Nc) ? c : (Nc - 1);
    bo0[t] = kh * ldb + cc;                  // row k0+kh
    bo1[t] = bo0[t] + ldb;                   // row k0+kh+1
  }
  const size_t aoff = (size_t)(row0 + am) * lda + kh;  // per-lane A offset

  v8f acc[NT];
#pragma unroll
  for (int t = 0; t < NT; ++t) acc[t] = v8f{};

  size_t kb = 0;                              // uniform k-base into B (floats)
  for (int k0 = 0; k0 < K; k0 += 4, kb += (size_t)4 * ldb) {
    const v2f a = load2(A + aoff + k0);
#pragma unroll
    for (int t = 0; t < NT; ++t) {
      v2f b;
      b.x = B[kb + bo0[t]] * cm[t];
      b.y = B[kb + bo1[t]] * cm[t];
      acc[t] = wmma4(a, b, acc[t]);
    }
  }

  const int rbase = row0 + ((lane >> 4) << 3);
#pragma unroll
  for (int t = 0; t < NT; ++t) {
    if (cols[t] < Nc) {
      const float bv = bias ? bias[cols[t]] : 0.f;
#pragma unroll
      for (int j = 0; j < 8; ++j)
        C[(size_t)(rbase + j) * ldc + cols[t]] = acc[t][j] + bv;
    }
  }
}

// ---------------------------------------------------------------------------
// Apply rigid frames: raw projection [N, H, 3, P] -> global points
// [N, H*P, 3]:  g = R p + t.   One thread per (n,h,p).
// ---------------------------------------------------------------------------
__global__ void frames_kernel(const float* __restrict__ raw,
                              const float* __restrict__ rot,
                              const float* __restrict__ trans,
                              float* __restrict__ outg, int P) {
  const int idx = blockIdx.x * blockDim.x + threadIdx.x;
  if (idx >= N * H * 8) return;               // launched with exact P grid
  const int p = idx % P;
  const int h = (idx / P) % H;
  const int n = idx / (P * H);
  if (n >= N) return;
  const float* r  = rot + (size_t)n * 9;
  const float* rw = raw + (size_t)n * (H * 3 * P) + h * 3 * P;
  const float px = rw[0 * P + p], py = rw[1 * P + p], pz = rw[2 * P + p];
  float* o = outg + ((size_t)n * (H * P) + h * P + p) * 3;
  o[0] = r[0] * px + r[1] * py + r[2] * pz + trans[n * 3 + 0];
  o[1] = r[3] * px + r[4] * py + r[5] * pz + trans[n * 3 + 1];
  o[2] = r[6] * px + r[7] * py + r[8] * pz + trans[n * 3 + 2];
}

// Per-(n,h) squared-norm sums over the PQK points (12 contiguous f32).
__global__ void pointnorm_kernel(const float* __restrict__ pg,
                                 float* __restrict__ nrm) {
  const int idx = blockIdx.x * blockDim.x + threadIdx.x;
  if (idx >= N * H) return;
  const float* s = pg + (size_t)(idx / H) * (H * PQK * 3) + (idx % H) * (PQK * 3);
  float acc = 0.f;
#pragma unroll
  for (int t = 0; t < PQK * 3; ++t) acc += s[t] * s[t];
  nrm[idx] = acc;
}

// Pack concat(v, value-points) into a zero-padded [N, H, 48] buffer so the
// attention*value GEMM streams contiguous, unconditional B fragments.
__global__ void packv_kernel(const float* __restrict__ v,    // [N, H*16]
                             const float* __restrict__ vpg,  // [N, H*24]
                             float* __restrict__ vcat) {     // [N, H*48]
  const int idx = blockIdx.x * blockDim.x + threadIdx.x;
  if (idx >= N * H * VCW) return;
  const int c = idx % VCW;
  const int h = (idx / VCW) % H;
  const int n = idx / (VCW * H);
  float val = 0.f;
  if (c < SV)           val = v[(size_t)n * (H * SV) + h * SV + c];
  else if (c < SV + 24) val = vpg[(size_t)n * (H * 24) + h * 24 + (c - SV)];
  vcat[idx] = val;
}

// ---------------------------------------------------------------------------
// Logits: acc = q·k + pw*(qp·kp) via one WMMA chain with K = 16 + 12 = 28.
// The per-K-block source switch (scalar vs point channels) is uniform, so the
// loop is split in two — no per-lane control flow anywhere in the hot path.
// Epilogue adds -0.5*pw*(|qp|^2+|kp|^2), the pre-filled 2D bias, mask, scale.
// ---------------------------------------------------------------------------
__global__ void logits_kernel(const float* __restrict__ qm,   // [N, H*16]
                              const float* __restrict__ km,   // [N, H*16]
                              const float* __restrict__ qpg,  // [N, H*12]
                              const float* __restrict__ kpg,  // [N, H*12]
                              const float* __restrict__ qn,   // [N, H]
                              const float* __restrict__ kn,   // [N, H]
                              const float* __restrict__ tw,   // [H]
                              const float* __restrict__ smask,// [N]
                              float* __restrict__ logits) {   // [N,N,H]
  const int wave = (blockIdx.x * blockDim.x + threadIdx.x) >> 5;
  const int lane = threadIdx.x & 31;
  const int qt16 = N >> 4;
  if (wave >= qt16 * qt16 * H) return;
  const int h   = wave / (qt16 * qt16);
  const int rem = wave % (qt16 * qt16);
  const int q0  = (rem % qt16) << 4;
  const int k0v = (rem / qt16) << 4;

  const float pw = sqrtf(2.f / (9.f * (float)PQK)) * softplusf(tw[h]);
  const int ml  = lane & 15;
  const int khf = (lane >> 4) << 1;

  const float* qrow  = qm  + (size_t)(q0  + ml) * (H * SQK) + h * SQK + khf;
  const float* krow  = km  + (size_t)(k0v + ml) * (H * SQK) + h * SQK + khf;
  const float* qprow = qpg + (size_t)(q0  + ml) * (H * PQK * 3) + h * (PQK * 3) + khf;
  const float* kprow = kpg + (size_t)(k0v + ml) * (H * PQK * 3) + h * (PQK * 3) + khf;

  v8f acc = {};
#pragma unroll
  for (int c0 = 0; c0 < SQK; c0 += 4)         // scalar q/k channels
    acc = wmma4(load2(qrow + c0) * 0.25f, load2(krow + c0), acc);
#pragma unroll
  for (int c0 = 0; c0 < PQK * 3; c0 += 4)     // point channels, pw folded in A
    acc = wmma4(load2(qprow + c0) * pw, load2(kprow + c0), acc);

  const int kcol  = k0v + (lane & 15);
  const int rbase = q0 + ((lane >> 4) << 3);
  const float smk   = smask[kcol];
  const float knv   = kn[kcol * H + h];
#pragma unroll
  for (int j = 0; j < 8; ++j) {
    const int qrowi = rbase + j;
    const size_t idx = ((size_t)qrowi * N + kcol) * H + h;
    float l = logits[idx] + acc[j] - 0.5f * pw * (qn[qrowi * H + h] + knv);
    l -= 1e5f * (1.f - smask[qrowi] * smk);
    logits[idx] = l * INV_SQRT3;
  }
}

// Softmax over the key axis (stride H).  One wave32 per (q,h) row.
__global__ void softmax_kernel(float* __restrict__ logits) {
  const int wave = (blockIdx.x * blockDim.x + threadIdx.x) >> 5;
  const int lane = threadIdx.x & 31;
  if (wave >= N * H) return;
  float* base = logits + (size_t)(wave / H) * N * H + (wave % H);

  float mx = -3.4e38f;
  for (int k = lane; k < N; k += 32) mx = fmaxf(mx, base[(size_t)k * H]);
#pragma unroll
  for (int o = 16; o > 0; o >>= 1) mx = fmaxf(mx, __shfl_xor(mx, o, 32));
  float sum = 0.f;
  for (int k = lane; k < N; k += 32) {
    const float e = expf(base[(size_t)k * H] - mx);
    base[(size_t)k * H] = e;
    sum += e;
  }
#pragma unroll
  for (int o = 16; o > 0; o >>= 1) sum += __shfl_xor(sum, o, 32);
  const float inv = 1.f / sum;
  for (int k = lane; k < N; k += 32) base[(size_t)k * H] *= inv;
}

// ---------------------------------------------------------------------------
// out[q,c] = sum_k attn[q,k,h] * vcat[k,h,c], c in [0,48).  One wave per
// (h, q-tile) carries all 3 column subtiles -> one A fragment per 3 WMMAs.
// Column routing (res_scalar vs rpg vs pad) happens only in the epilogue.
// ---------------------------------------------------------------------------
__global__ void attn_v_kernel(const float* __restrict__ attn, // [N,N,H]
                              const float* __restrict__ vcat, // [N, H*48]
                              float* __restrict__ finalbuf,   // [N, FIN]
                              float* __restrict__ rpg) {      // [N, H*24]
  const int wave = (blockIdx.x * blockDim.x + threadIdx.x) >> 5;
  const int lane = threadIdx.x & 31;
  const int mtiles = N >> 4;
  if (wave >= H * mtiles) return;
  const int h  = wave / mtiles;
  const int q0 = (wave % mtiles) << 4;
  const int am  = lane & 15;
  const int khf = (lane >> 4) << 1;

  const size_t aoff = ((size_t)(q0 + am) * N + khf) * H + h;  // per-lane, +k*H
  const int    boff = khf * (H * VCW) + h * VCW + (lane & 15);

  v8f acc[3];
#pragma unroll
  for (int t = 0; t < 3; ++t) acc[t] = v8f{};

  for (int k0 = 0; k0 < N; k0 += 4) {
    v2f a;
    const size_t ab = aoff + (size_t)k0 * H;
    a.x = attn[ab];
    a.y = attn[ab + H];
    const float* bp = vcat + (size_t)k0 * (H * VCW) + boff;
#pragma unroll
    for (int t = 0; t < 3; ++t) {
      v2f b;
      b.x = bp[t * 16];
      b.y = bp[t * 16 + H * VCW];
      acc[t] = wmma4(a, b, acc[t]);
    }
  }

  const int rbase = q0 + ((lane >> 4) << 3);
#pragma unroll
  for (int t = 0; t < 3; ++t) {
    const int cn = t * 16 + (lane & 15);
#pragma unroll
    for (int j = 0; j < 8; ++j) {
      const int q = rbase + j;
      if (cn < SV)
        finalbuf[(size_t)q * FIN + h * SV + cn] = acc[t][j];
      else if (cn < SV + 24)
        rpg[(size_t)q * (H * 24) + h * 24 + (cn - SV)] = acc[t][j];
    }
  }
}

// ---------------------------------------------------------------------------
// r2d[q,h,c] = sum_k attn[q,k,h] * act_2d[q,k,c].  One workgroup (4 waves,
// 2 column subtiles each = 128 cols) per q; the 393 KB act_2d slab for this
// q is shared through cache and prefetched ahead (global_prefetch_b8).
// A rows are heads (12 padded to 16 via clamp + 0/1 mask -> no exec toggles).
// ---------------------------------------------------------------------------
__global__ void r2d_kernel(const float* __restrict__ attn,
                           const float* __restrict__ act2d,
                           float* __restrict__ finalbuf) {
  const int q    = blockIdx.x;
  const int wv   = threadIdx.x >> 5;          // 0..3
  const int lane = threadIdx.x & 31;
  const int am   = lane & 15;
  const int khf  = (lane >> 4) << 1;
  const float hm = (am < H) ? 1.f : 0.f;
  const int  hc  = (am < H) ? am : (H - 1);
  const int  c0  = (wv << 5) + (lane & 15);   // subtile0 col; subtile1 = +16

  const float* b2 = act2d + (size_t)q * N * C2;
  const size_t aoff = ((size_t)q * N + khf) * H + hc;

  v8f acc[2];
  acc[0] = v8f{}; acc[1] = v8f{};

  for (int k0 = 0; k0 < N; k0 += 4) {
    if (k0 + 32 < N)
      __builtin_prefetch(b2 + (size_t)(k0 + 32 + khf) * C2 + c0, 0, 1);
    v2f a;
    const size_t ab = aoff + (size_t)k0 * H;
    a.x = attn[ab] * hm;
    a.y = attn[ab + H] * hm;
    const float* bp = b2 + (size_t)(k0 + khf) * C2 + c0;
#pragma unroll
    for (int t = 0; t < 2; ++t) {
      v2f b;
      b.x = bp[t * 16];
      b.y = bp[t * 16 + C2];
      acc[t] = wmma4(a, b, acc[t]);
    }
  }

  const int hb = (lane >> 4) << 3;
#pragma unroll
  for (int t = 0; t < 2; ++t) {
#pragma unroll
    for (int j = 0; j < 8; ++j) {
      const int h = hb + j;
      if (h < H)
        finalbuf[(size_t)q * FIN + (H * SV + 4 * H * PV) + h * C2 + c0 + t * 16]
            = acc[t][j];
    }
  }
}

// Invert frames on attended value points, compute norms, scatter into the
// final concat layout [res_scalar | x | y | z | norm | r2d].
__global__ void finalize_kernel(const float* __restrict__ rpg,
                                const float* __restrict__ rot,
                                const float* __restrict__ trans,
                                float* __restrict__ finalbuf) {
  const int idx = blockIdx.x * blockDim.x + threadIdx.x;
  if (idx >= N * H * PV) return;
  const int hp = idx % (H * PV);
  const int n  = idx / (H * PV);
  const float* r = rot + (size_t)n * 9;
  const float* g = rpg + ((size_t)n * (H * PV) + hp) * 3;
  const float gx = g[0] - trans[n * 3 + 0];
  const float gy = g[1] - trans[n * 3 + 1];
  const float gz = g[2] - trans[n * 3 + 2];
  const float lx = r[0] * gx + r[3] * gy + r[6] * gz;   // R^T (g - t)
  const float ly = r[1] * gx + r[4] * gy + r[7] * gz;
  const float lz = r[2] * gx + r[5] * gy + r[8] * gz;
  float* fb = finalbuf + (size_t)n * FIN + H * SV;
  fb[0 * H * PV + hp] = lx;
  fb[1 * H * PV + hp] = ly;
  fb[2 * H * PV + hp] = lz;
  fb[3 * H * PV + hp] = sqrtf(lx * lx + ly * ly + lz * lz);
}

// ---------------------------------------------------------------------------
template <int NT>
static inline void gemm16(hipStream_t s, const float* A, const float* B,
                          const float* bias, float* C,
                          int M, int Nc, int K, int lda, int ldb, int ldc) {
  const int waves  = (M >> 4) * ((Nc + 16 * NT - 1) / (16 * NT));
  const int blocks = (waves * 32 + 255) / 256;
  gemm16_kernel<NT><<<blocks, 256, 0, s>>>(A, B, bias, C, M, Nc, K, lda, ldb, ldc);
}

extern "C" void kernel_launch(void* const* d_in, const int* in_sizes, int n_in,
                              void* d_out, int out_size, void* d_ws, size_t ws_size,
                              hipStream_t stream) {
  (void)in_sizes; (void)n_in; (void)out_size; (void)ws_size;
  const float* act   = (const float*)d_in[0];
  const float* act2d = (const float*)d_in[1];
  const float* smask = (const float*)d_in[2];
  const float* rot   = (const float*)d_in[3];
  const float* trans = (const float*)d_in[4];
  const float* wq    = (const float*)d_in[5];
  const float* wk    = (const float*)d_in[6];
  const float* wv    = (const float*)d_in[7];
  const float* wqp   = (const float*)d_in[8];
  const float* bqp   = (const float*)d_in[9];
  const float* wkp   = (const float*)d_in[10];
  const float* bkp   = (const float*)d_in[11];
  const float* wvp   = (const float*)d_in[12];
  const float* bvp   = (const float*)d_in[13];
  const float* w2d   = (const float*)d_in[14];
  const float* b2d   = (const float*)d_in[15];
  const float* wfin  = (const float*)d_in[16];
  const float* bfin  = (const float*)d_in[17];
  const float* tw    = (const float*)d_in[18];
  float* out = (float*)d_out;

  // Workspace carve-up (floats); total ~43 MB.
  float* ws = (float*)d_ws;
  size_t off = 0;
  auto alloc = [&](size_t nf) { float* p = ws + off; off += nf; return p; };
  float* qbuf   = alloc((size_t)N * H * SQK);
  float* kbuf   = alloc((size_t)N * H * SQK);
  float* vbuf   = alloc((size_t)N * H * SV);
  float* qpraw  = alloc((size_t)N * H * PQK * 3);
  float* kpraw  = alloc((size_t)N * H * PQK * 3);
  float* vpraw  = alloc((size_t)N * H * PV * 3);
  float* qpg    = alloc((size_t)N * H * PQK * 3);
  float* kpg    = alloc((size_t)N * H * PQK * 3);
  float* vpg    = alloc((size_t)N * H * PV * 3);
  float* qn     = alloc((size_t)N * H);
  float* kn     = alloc((size_t)N * H);
  float* rpg    = alloc((size_t)N * H * PV * 3);
  float* vcat   = alloc((size_t)N * H * VCW);
  float* finalb = alloc((size_t)N * FIN);
  float* logits = alloc((size_t)N * N * H);   // 28.3 MB

  // 1. Scalar + point projections (WMMA GEMMs, K = C1 = 384).
  gemm16<4>(stream, act, wq,  nullptr, qbuf,  N, H * SQK,     C1, C1, H * SQK,     H * SQK);
  gemm16<4>(stream, act, wk,  nullptr, kbuf,  N, H * SQK,     C1, C1, H * SQK,     H * SQK);
  gemm16<4>(stream, act, wv,  nullptr, vbuf,  N, H * SV,      C1, C1, H * SV,      H * SV);
  gemm16<3>(stream, act, wqp, bqp,     qpraw, N, H * PQK * 3, C1, C1, H * PQK * 3, H * PQK * 3);
  gemm16<3>(stream, act, wkp, bkp,     kpraw, N, H * PQK * 3, C1, C1, H * PQK * 3, H * PQK * 3);
  gemm16<3>(stream, act, wvp, bvp,     vpraw, N, H * PV * 3,  C1, C1, H * PV * 3,  H * PV * 3);

  // 2. Apply rigid frames; point norm sums; pack values.
  frames_kernel<<<(N * H * PQK + 255) / 256, 256, 0, stream>>>(qpraw, rot, trans, qpg, PQK);
  frames_kernel<<<(N * H * PQK + 255) / 256, 256, 0, stream>>>(kpraw, rot, trans, kpg, PQK);
  frames_kernel<<<(N * H * PV  + 255) / 256, 256, 0, stream>>>(vpraw, rot, trans, vpg, PV);
  pointnorm_kernel<<<(N * H + 255) / 256, 256, 0, stream>>>(qpg, qn);
  pointnorm_kernel<<<(N * H + 255) / 256, 256, 0, stream>>>(kpg, kn);
  packv_kernel<<<(N * H * VCW + 255) / 256, 256, 0, stream>>>(vbuf, vpg, vcat);

  // 3. 2D bias: logits[q,k,h] = act_2d[q,k,:] @ w2d + b2d  (M = N*N rows).
  gemm16<1>(stream, act2d, w2d, b2d, logits, N * N, H, C2, C2, H, H);

  // 4. Attention logits (WMMA, K = 28) + norm/mask/scale epilogue.
  {
    const int waves = (N / 16) * (N / 16) * H;
    logits_kernel<<<(waves * 32 + 255) / 256, 256, 0, stream>>>(
        qbuf, kbuf, qpg, kpg, qn, kn, tw, smask, logits);
  }

  // 5. Softmax over keys (one wave per (q,h)).
  softmax_kernel<<<(N * H * 32 + 255) / 256, 256, 0, stream>>>(logits);

  // 6. Attention applied to packed scalar values + value points.
  {
    const int waves = H * (N / 16);
    attn_v_kernel<<<(waves * 32 + 255) / 256, 256, 0, stream>>>(
        logits, vcat, finalb, rpg);
  }

  // 7. Attention applied to act_2d (dominant 302 MB HBM stream).
  r2d_kernel<<<N, 128, 0, stream>>>(logits, act2d, finalb);

  // 8. Invert frames, norms, concat.
  finalize_kernel<<<(N * H * PV + 255) / 256, 256, 0, stream>>>(rpg, rot, trans, finalb);

  // 9. Final projection: [N, 2112] @ [2112, 384] + bfin.
  gemm16<4>(stream, finalb, wfin, bfin, out, N, C1, FIN, FIN, C1, C1);
}